// VectorQuantizer_44702019617299
// MI455X (gfx1250) — compile-verified
//
#include <hip/hip_runtime.h>
#include <hip/hip_bf16.h>

typedef __attribute__((ext_vector_type(2))) float v2f;
typedef __attribute__((ext_vector_type(8))) float v8f;

#define DIMD   256
#define KCODES 8192
#define MTOT   32768
#define TM     64      // rows per block (4 waves x 16)
#define TN     16      // codes per chunk
#define PITCH  260     // LDS row pitch in floats (conflict-free b64 reads)
#define NCHUNK (KCODES / TN)   // 512
#define CHUNK_BYTES (TN * PITCH * 4)

// CDNA5 async global->LDS copy: 16B per lane, tracked with ASYNCcnt.
__device__ __forceinline__ void async_ld16(const float* g, unsigned lds_off) {
    unsigned long long ga = (unsigned long long)g;
    asm volatile("global_load_async_to_lds_b128 %0, %1, off"
                 :: "v"(lds_off), "v"(ga)
                 : "memory");
}

// --------------------------------------------------------------------------
// Kernel 1: e_sq[k] = sum_d codebook[k][d]^2   (8 threads per row, coalesced)
// --------------------------------------------------------------------------
__global__ __launch_bounds__(256) void vq_esq_kernel(const float* __restrict__ cb,
                                                     float* __restrict__ esq) {
    const int t   = threadIdx.x;
    const int row = blockIdx.x * 32 + (t >> 3);
    const int sub = t & 7;
    float s = 0.0f;
#pragma unroll
    for (int j = 0; j < 8; ++j) {
        const float4 v = *(const float4*)(cb + row * DIMD + sub * 4 + j * 32);
        s += v.x * v.x + v.y * v.y + v.z * v.z + v.w * v.w;
    }
    s += __shfl_xor(s, 1, 32);
    s += __shfl_xor(s, 2, 32);
    s += __shfl_xor(s, 4, 32);
    if (sub == 0) esq[row] = s;
}

// --------------------------------------------------------------------------
// Kernel 2: main — WMMA distance GEMM + argmin + gather + partial loss
// --------------------------------------------------------------------------
__global__ __launch_bounds__(128) void vq_main_kernel(const float* __restrict__ z,
                                                      const float* __restrict__ cb,
                                                      const float* __restrict__ esq,
                                                      float* __restrict__ zq_out,
                                                      float* __restrict__ idx_out,
                                                      float* __restrict__ partials) {
    __shared__ float sB[3][TN * PITCH];   // triple-buffered codebook chunks
    __shared__ int   sIdx[TM];            // argmin per block row
    __shared__ float sRed[128];           // loss reduction

    const int tid  = threadIdx.x;
    const int lane = tid & 31;
    const int wave = tid >> 5;
    const int m0   = blockIdx.x * TM;
    const int lm   = lane & 15;        // M (for A) / N (for B) within tile
    const int hi   = lane >> 4;        // K-half select

    // LDS byte base of sB (generic LDS pointer: low 32 bits = LDS offset)
    const unsigned sB_base = (unsigned)(unsigned long long)&sB[0][0];

    // ---- issue one chunk's async copies (8 x b128 per thread)
    auto issue_chunk = [&](int nc, int buf) {
        const float* src = cb + nc * TN * DIMD;
        const unsigned dst = sB_base + (unsigned)buf * CHUNK_BYTES;
#pragma unroll
        for (int j = 0; j < 8; ++j) {
            const int e4 = tid + j * 128;     // float4 index 0..1023
            const int r  = e4 >> 6;           // code row 0..15
            const int c4 = e4 & 63;           // float4 column
            async_ld16(src + r * DIMD + c4 * 4,
                       dst + (unsigned)(r * (PITCH * 4) + c4 * 16));
        }
    };

    // prologue: chunks 0 and 1 in flight
    issue_chunk(0, 0);
    issue_chunk(1, 1);

    // ---- load A tile: 16 rows x 256, f32 WMMA A layout:
    //      lanes 0-15 hold K={4kk, 4kk+1}; lanes 16-31 hold K={4kk+2, 4kk+3}
    const int rowA = m0 + wave * 16 + lm;
    v2f a[64];
#pragma unroll
    for (int kk = 0; kk < 64; ++kk) {
        a[kk] = *(const v2f*)(z + rowA * DIMD + kk * 4 + hi * 2);
    }

    unsigned long long run[8];
#pragma unroll
    for (int v = 0; v < 8; ++v) run[v] = ~0ull;

    int bufc = 0;   // nc % 3, maintained incrementally (no division)
    for (int nc = 0; nc < NCHUNK; ++nc) {
        // chunk nc complete (async loads finish in order); nc+1 stays in flight
        if (nc + 1 < NCHUNK) {
            asm volatile("s_wait_asynccnt 0x8" ::: "memory");
        } else {
            asm volatile("s_wait_asynccnt 0x0" ::: "memory");
        }
        __syncthreads();   // buffer bufc staged block-wide; prior reads done

        if (nc + 2 < NCHUNK) {
            const int bufn = (bufc >= 1) ? (bufc - 1) : (bufc + 2);  // (nc+2)%3
            issue_chunk(nc + 2, bufn);
        }

        const float* sBlane = &sB[bufc][0] + lm * PITCH + hi * 2;

        // ---- 16x16 tile of z @ cb^T: 64 f32 WMMAs, B fragments pipelined 4 deep
        v8f c0 = {};
        v8f c1 = {};
        v2f p0 = *(const v2f*)(sBlane + 0);
        v2f p1 = *(const v2f*)(sBlane + 4);
        v2f p2 = *(const v2f*)(sBlane + 8);
        v2f p3 = *(const v2f*)(sBlane + 12);
#pragma unroll
        for (int kk = 0; kk < 64; kk += 4) {
            v2f n0, n1, n2, n3;
            if (kk + 4 < 64) {
                n0 = *(const v2f*)(sBlane + (kk + 4) * 4);
                n1 = *(const v2f*)(sBlane + (kk + 5) * 4);
                n2 = *(const v2f*)(sBlane + (kk + 6) * 4);
                n3 = *(const v2f*)(sBlane + (kk + 7) * 4);
            }
            c0 = __builtin_amdgcn_wmma_f32_16x16x4_f32(false, a[kk + 0], false, p0,
                                                       (short)0, c0, false, false);
            c1 = __builtin_amdgcn_wmma_f32_16x16x4_f32(false, a[kk + 1], false, p1,
                                                       (short)0, c1, false, false);
            c0 = __builtin_amdgcn_wmma_f32_16x16x4_f32(false, a[kk + 2], false, p2,
                                                       (short)0, c0, false, false);
            c1 = __builtin_amdgcn_wmma_f32_16x16x4_f32(false, a[kk + 3], false, p3,
                                                       (short)0, c1, false, false);
            if (kk + 4 < 64) { p0 = n0; p1 = n1; p2 = n2; p3 = n3; }
        }

        // ---- fold esq, update running argmin (packed sortable key | index)
        const int   nIdx = nc * TN + lm;
        const float es   = esq[nIdx];
#pragma unroll
        for (int v = 0; v < 8; ++v) {
            const float ze  = c0[v] + c1[v];
            const float key = es - 2.0f * ze;                 // z_sq omitted (row const)
            unsigned int su = __float_as_uint(key);
            su = (su & 0x80000000u) ? ~su : (su | 0x80000000u);
            const unsigned long long pk =
                ((unsigned long long)su << 32) | (unsigned int)nIdx;
            run[v] = (pk < run[v]) ? pk : run[v];
        }

        bufc = (bufc == 2) ? 0 : (bufc + 1);
    }

    // ---- reduce argmin across the 16 N-lanes of each half-wave
#pragma unroll
    for (int v = 0; v < 8; ++v) {
        unsigned long long pk = run[v];
#pragma unroll
        for (int off = 1; off < 16; off <<= 1) {
            const unsigned long long o = __shfl_xor(pk, off, 32);
            pk = (o < pk) ? o : pk;
        }
        if (lm == 0) {
            const int row = wave * 16 + v + 8 * hi;   // row within block
            const int ci  = (int)(pk & 0xFFFFFFFFull);
            sIdx[row] = ci;
            idx_out[m0 + row] = (float)ci;
        }
    }
    __syncthreads();

    // ---- gather codebook rows, emit z_q_st = z_e + (z_q - z_e), accumulate loss
    float acc = 0.0f;
    for (int e = tid; e < TM * DIMD; e += 128) {
        const int r   = e >> 8;
        const int col = e & 255;
        const int ci  = sIdx[r];
        const float q  = cb[ci * DIMD + col];
        const float ze = z[(m0 + r) * DIMD + col];
        const float d  = q - ze;
        zq_out[(m0 + r) * DIMD + col] = ze + d;   // mirrors reference STE arithmetic
        acc += d * d;
    }
    sRed[tid] = acc;
    __syncthreads();
#pragma unroll
    for (int s = 64; s > 0; s >>= 1) {
        if (tid < s) sRed[tid] += sRed[tid + s];
        __syncthreads();
    }
    if (tid == 0) partials[blockIdx.x] = sRed[0];
}

// --------------------------------------------------------------------------
// Kernel 3: deterministic final loss reduction
// --------------------------------------------------------------------------
__global__ __launch_bounds__(512) void vq_fin_kernel(const float* __restrict__ partials,
                                                     float* __restrict__ loss) {
    __shared__ float s[512];
    const int t = threadIdx.x;
    s[t] = partials[t];          // exactly 512 partials
    __syncthreads();
#pragma unroll
    for (int off = 256; off > 0; off >>= 1) {
        if (t < off) s[t] += s[t + off];
        __syncthreads();
    }
    if (t == 0) loss[0] = s[0] * (1.25f / 8388608.0f);
}

// --------------------------------------------------------------------------
extern "C" void kernel_launch(void* const* d_in, const int* in_sizes, int n_in,
                              void* d_out, int out_size, void* d_ws, size_t ws_size,
                              hipStream_t stream) {
    const float* z  = (const float*)d_in[0];   // [32,1024,256] f32
    const float* cb = (const float*)d_in[1];   // [8192,256]   f32

    float* out  = (float*)d_out;
    float* zq   = out;                  // 8388608 floats
    float* loss = out + 8388608;        // 1 float
    float* idxf = out + 8388609;        // 32768 floats

    float* esq      = (float*)d_ws;     // 8192 floats
    float* partials = esq + KCODES;     // 512 floats

    vq_esq_kernel<<<KCODES / 32, 256, 0, stream>>>(cb, esq);
    vq_main_kernel<<<MTOT / TM, 128, 0, stream>>>(z, cb, esq, zq, idxf, partials);
    vq_fin_kernel<<<1, 512, 0, stream>>>(partials, loss);
}